// SMPL_42245298324239
// MI455X (gfx1250) — compile-verified
//
#include <hip/hip_runtime.h>
#include <math.h>

// Only check for the device builtin when actually compiling for the device:
// __has_builtin(__builtin_amdgcn_*) is false during hipcc's host (x86) pass.
#if defined(__HIP_DEVICE_COMPILE__)
#if !__has_builtin(__builtin_amdgcn_wmma_f32_16x16x4_f32)
#error "fp32 WMMA builtin __builtin_amdgcn_wmma_f32_16x16x4_f32 not available on gfx1250 device pass"
#endif
#endif

typedef __attribute__((ext_vector_type(2))) float v2f;
typedef __attribute__((ext_vector_type(8))) float v8f;

#define BATCH 128
#define NV    6890
#define NTOT  (NV * 3)        // 20670
#define NJ    24
#define NB    10
#define NP    207
#define NPP   208             // pose_feature row stride (padded, last elem = 0)
#define JOUT  43

__device__ constexpr int kParents[NJ] = {0, 0, 0, 0, 1, 2, 3, 4, 5, 6, 7, 8,
                                         9, 9, 9, 12, 13, 14, 16, 17, 18, 19, 20, 21};

static __device__ inline v8f wmma_f32(v2f a, v2f b, v8f c) {
  return __builtin_amdgcn_wmma_f32_16x16x4_f32(false, a, false, b, (short)0, c,
                                               false, false);
}

// ---------------------------------------------------------------------------
// K1: Rodrigues -> Rs (B*24*9) and pose_feature (B*NPP, padded with one zero)
// ---------------------------------------------------------------------------
__global__ __launch_bounds__(256) void rodrigues_kernel(
    const float* __restrict__ theta, float* __restrict__ Rs,
    float* __restrict__ pf) {
  int idx = blockIdx.x * blockDim.x + threadIdx.x;
  if (idx >= BATCH * NJ) return;
  int b = idx / NJ, j = idx % NJ;
  float tx = theta[b * (NJ * 3) + j * 3 + 0];
  float ty = theta[b * (NJ * 3) + j * 3 + 1];
  float tz = theta[b * (NJ * 3) + j * 3 + 2];
  float ax = tx + 1e-8f, ay = ty + 1e-8f, az = tz + 1e-8f;
  float angle = sqrtf(ax * ax + ay * ay + az * az);
  float inv = 1.0f / angle;
  float rx = tx * inv, ry = ty * inv, rz = tz * inv;
  float c = cosf(angle), s = sinf(angle), ic = 1.0f - c;
  float R[9];
  R[0] = c + ic * rx * rx;  R[1] = ic * rx * ry - s * rz;  R[2] = ic * rx * rz + s * ry;
  R[3] = ic * ry * rx + s * rz;  R[4] = c + ic * ry * ry;  R[5] = ic * ry * rz - s * rx;
  R[6] = ic * rz * rx - s * ry;  R[7] = ic * rz * ry + s * rx;  R[8] = c + ic * rz * rz;
#pragma unroll
  for (int k = 0; k < 9; ++k) Rs[idx * 9 + k] = R[k];
  if (j >= 1) {
#pragma unroll
    for (int k = 0; k < 9; ++k) {
      float eye = (k == 0 || k == 4 || k == 8) ? 1.0f : 0.0f;
      pf[b * NPP + (j - 1) * 9 + k] = R[k] - eye;
    }
  } else {
    pf[b * NPP + NP] = 0.0f;  // zero the pad element
  }
}

// ---------------------------------------------------------------------------
// K2: fused shape+pose blend GEMM via V_WMMA_F32_16X16X4_F32.
//   v_shaped = beta @ shapedirs + v_template       (snapshot after K=10)
//   v_posed  = v_shaped + pose_feature @ posedirs  (continue same accumulator)
// One wave per 16x16 tile; edge tiles overlap (duplicate identical writes),
// so every load/store is unconditional -> no EXEC-diverged loads.
// ---------------------------------------------------------------------------
__global__ __launch_bounds__(32) void blend_gemm_kernel(
    const float* __restrict__ beta,       // 128 x 10
    const float* __restrict__ pf,         // 128 x NPP (padded)
    const float* __restrict__ shapedirs,  // 10 x 20670
    const float* __restrict__ posedirs,   // 207 x 20670
    const float* __restrict__ v_template, // 20670
    float* __restrict__ v_shaped,         // 128 x 20670
    float* __restrict__ v_posed) {        // 128 x 20670
  const int lane = threadIdx.x & 31;
  const int half = lane >> 4;
  const int lr = lane & 15;
  int nbase = blockIdx.x * 16;
  if (nbase > NTOT - 16) nbase = NTOT - 16;  // overlapping edge tile (uniform)
  const int mtile = blockIdx.y;
  const int ncol = nbase + lr;               // always in-bounds
  const int arow = mtile * 16 + lr;          // batch row (M = lane%16), <128

  v8f acc = {};

  // ---- shape phase: K = 10 -> two full chunks + branchless tail ----
#pragma unroll
  for (int k0 = 0; k0 < 8; k0 += 4) {
    int ka = k0 + 2 * half;  // even, <= 7
    v2f a = *(const v2f*)(beta + arow * NB + ka);
    v2f bm;
    bm.x = shapedirs[ka * NTOT + ncol];
    bm.y = shapedirs[(ka + 1) * NTOT + ncol];
    acc = wmma_f32(a, bm, acc);
  }
  {
    // tail chunk: ka in {8,10}; half==1 lanes are fully OOB in K -> A = 0.
    float t0 = beta[arow * NB + 8];
    float t1 = beta[arow * NB + 9];
    v2f a, bm;
    a.x = half ? 0.0f : t0;
    a.y = half ? 0.0f : t1;
    bm.x = shapedirs[(half ? 8 : 8) * NTOT + ncol];   // clamped row, A=0 kills it
    bm.y = shapedirs[9 * NTOT + ncol];
    acc = wmma_f32(a, bm, acc);
  }

  const float vt = v_template[ncol];
#pragma unroll
  for (int i = 0; i < 8; ++i) {
    int row = mtile * 16 + i + 8 * half;  // C layout: vgpr i -> M = i + 8*(lane/16)
    v_shaped[row * NTOT + ncol] = acc[i] + vt;
  }

  // ---- pose phase: K = 207 (pf padded to 208) ----
  // 51 fully-valid chunks (ka+1 <= 203), then one tail chunk where the only
  // OOB element is pf's zero pad (A=0 nullifies the clamped B row).
#pragma unroll 4
  for (int k0 = 0; k0 < 204; k0 += 4) {
    int ka = k0 + 2 * half;  // even, <= 202
    v2f a = *(const v2f*)(pf + arow * NPP + ka);
    v2f bm;
    bm.x = posedirs[ka * NTOT + ncol];
    bm.y = posedirs[(ka + 1) * NTOT + ncol];
    if (k0 + 20 < 204)  // uniform predicate -> emits global_prefetch_b8
      __builtin_prefetch(&posedirs[(k0 + 20) * NTOT + ncol], 0, 1);
    acc = wmma_f32(a, bm, acc);
  }
  {
    int ka = 204 + 2 * half;                  // 204 or 206
    int r1 = (ka + 1 < NP) ? ka + 1 : NP - 1; // clamp row 207 -> 206
    v2f a = *(const v2f*)(pf + arow * NPP + ka);  // pf[...,207] == 0 pad
    v2f bm;
    bm.x = posedirs[ka * NTOT + ncol];
    bm.y = posedirs[r1 * NTOT + ncol];
    acc = wmma_f32(a, bm, acc);
  }

#pragma unroll
  for (int i = 0; i < 8; ++i) {
    int row = mtile * 16 + i + 8 * half;
    v_posed[row * NTOT + ncol] = acc[i] + vt;
  }
}

// ---------------------------------------------------------------------------
// K3/K6: regression out[b,j,c] = sum_v reg[j,v] * src[b,v,c]. Block per (b,j).
// ---------------------------------------------------------------------------
struct F3 { float x, y, z; };

__global__ __launch_bounds__(256) void regress_kernel(
    const float* __restrict__ reg,  // nj x V
    const float* __restrict__ src,  // B x V x 3
    float* __restrict__ out,        // B x nj x 3
    int nj) {
  __shared__ float s0[256], s1[256], s2[256];
  const int tid = threadIdx.x;
  const int b = blockIdx.x / nj;
  const int j = blockIdx.x % nj;
  float a0 = 0.0f, a1 = 0.0f, a2 = 0.0f;
  const float* rj = reg + (size_t)j * NV;
  const F3* sb = (const F3*)(src + (size_t)b * NTOT);
  for (int v = tid; v < NV; v += 256) {
    float w = rj[v];
    F3 p = sb[v];
    a0 += w * p.x;
    a1 += w * p.y;
    a2 += w * p.z;
  }
  s0[tid] = a0; s1[tid] = a1; s2[tid] = a2;
  __syncthreads();
  for (int off = 128; off > 0; off >>= 1) {
    if (tid < off) {
      s0[tid] += s0[tid + off];
      s1[tid] += s1[tid + off];
      s2[tid] += s2[tid + off];
    }
    __syncthreads();
  }
  if (tid == 0) {
    out[(b * nj + j) * 3 + 0] = s0[0];
    out[(b * nj + j) * 3 + 1] = s1[0];
    out[(b * nj + j) * 3 + 2] = s2[0];
  }
}

// ---------------------------------------------------------------------------
// K4: kinematic chain. One thread per batch; fully unrolled with constexpr
// parents. Writes A as B x 24 x 16 (4x4 row-major, bottom row [0,0,0,1]),
// with translation column already adjusted:  t' = t_acc - R_acc @ J_orig.
// ---------------------------------------------------------------------------
__global__ __launch_bounds__(128) void chain_kernel(
    const float* __restrict__ Rs,   // B x 24 x 9
    const float* __restrict__ Jw,   // B x 24 x 3
    float* __restrict__ Aw) {       // B x 24 x 16
  int b = blockIdx.x * blockDim.x + threadIdx.x;
  if (b >= BATCH) return;
  float R[NJ][9], t[NJ][3];
  // root: Rs[:,0] @ diag(1,-1,-1)  (negate columns 1,2)
#pragma unroll
  for (int k = 0; k < 9; ++k) R[0][k] = Rs[(b * NJ + 0) * 9 + k];
  R[0][1] = -R[0][1]; R[0][2] = -R[0][2];
  R[0][4] = -R[0][4]; R[0][5] = -R[0][5];
  R[0][7] = -R[0][7]; R[0][8] = -R[0][8];
#pragma unroll
  for (int c = 0; c < 3; ++c) t[0][c] = Jw[(b * NJ + 0) * 3 + c];

#pragma unroll
  for (int i = 1; i < NJ; ++i) {
    const int p = kParents[i];
    float Ri[9], ji[3];
#pragma unroll
    for (int k = 0; k < 9; ++k) Ri[k] = Rs[(b * NJ + i) * 9 + k];
#pragma unroll
    for (int c = 0; c < 3; ++c)
      ji[c] = Jw[(b * NJ + i) * 3 + c] - Jw[(b * NJ + p) * 3 + c];
#pragma unroll
    for (int r = 0; r < 3; ++r) {
#pragma unroll
      for (int c = 0; c < 3; ++c)
        R[i][r * 3 + c] = R[p][r * 3 + 0] * Ri[0 * 3 + c] +
                          R[p][r * 3 + 1] * Ri[1 * 3 + c] +
                          R[p][r * 3 + 2] * Ri[2 * 3 + c];
      t[i][r] = R[p][r * 3 + 0] * ji[0] + R[p][r * 3 + 1] * ji[1] +
                R[p][r * 3 + 2] * ji[2] + t[p][r];
    }
  }

#pragma unroll
  for (int i = 0; i < NJ; ++i) {
    float jx = Jw[(b * NJ + i) * 3 + 0];
    float jy = Jw[(b * NJ + i) * 3 + 1];
    float jz = Jw[(b * NJ + i) * 3 + 2];
    float* o = Aw + (size_t)(b * NJ + i) * 16;
#pragma unroll
    for (int r = 0; r < 3; ++r) {
      o[r * 4 + 0] = R[i][r * 3 + 0];
      o[r * 4 + 1] = R[i][r * 3 + 1];
      o[r * 4 + 2] = R[i][r * 3 + 2];
      o[r * 4 + 3] = t[i][r] - (R[i][r * 3 + 0] * jx + R[i][r * 3 + 1] * jy +
                                R[i][r * 3 + 2] * jz);
    }
    o[12] = 0.0f; o[13] = 0.0f; o[14] = 0.0f; o[15] = 1.0f;
  }
}

// ---------------------------------------------------------------------------
// K5: LBS skinning via WMMA. Per (batch, 16-vertex tile):
//   T(16x16) = W_tile(16x24) @ A_b(24x16)   -> 6x v_wmma_f32_16x16x4_f32
// then lanes 0..15 apply their vertex's 3x4 affine to v_posed.
// Edge vertex tile overlaps (identical duplicate writes) -> no guards at all.
// ---------------------------------------------------------------------------
__global__ __launch_bounds__(32) void skin_kernel(
    const float* __restrict__ weights,  // V x 24
    const float* __restrict__ Aw,       // B x 24 x 16
    const float* __restrict__ v_posed,  // B x 20670
    float* __restrict__ verts) {        // B x V x 3 (d_out)
  __shared__ float Tl[256];
  const int lane = threadIdx.x & 31;
  const int half = lane >> 4;
  const int lr = lane & 15;
  int vbase = blockIdx.x * 16;
  if (vbase > NV - 16) vbase = NV - 16;  // overlapping edge tile (uniform)
  const int b = blockIdx.y;
  const int vrow = vbase + lr;           // always < NV
  const float* Ab = Aw + (size_t)b * NJ * 16;

  v8f acc = {};
#pragma unroll
  for (int k0 = 0; k0 < NJ; k0 += 4) {
    int ka = k0 + 2 * half;  // even, in [0,24)
    v2f a = *(const v2f*)(weights + vrow * NJ + ka);
    v2f bm;
    bm.x = Ab[ka * 16 + lr];
    bm.y = Ab[(ka + 1) * 16 + lr];
    acc = wmma_f32(a, bm, acc);
  }
#pragma unroll
  for (int i = 0; i < 8; ++i) Tl[(i + 8 * half) * 16 + lr] = acc[i];
  __syncthreads();

  if (lane < 16) {
    int v = vbase + lane;  // always < NV
    float x = v_posed[(size_t)b * NTOT + v * 3 + 0];
    float y = v_posed[(size_t)b * NTOT + v * 3 + 1];
    float z = v_posed[(size_t)b * NTOT + v * 3 + 2];
    const float* Tv = &Tl[lane * 16];
#pragma unroll
    for (int p = 0; p < 3; ++p) {
      verts[((size_t)b * NV + v) * 3 + p] =
          Tv[p * 4 + 0] * x + Tv[p * 4 + 1] * y + Tv[p * 4 + 2] * z +
          Tv[p * 4 + 3];
    }
  }
}

// ---------------------------------------------------------------------------
extern "C" void kernel_launch(void* const* d_in, const int* in_sizes, int n_in,
                              void* d_out, int out_size, void* d_ws,
                              size_t ws_size, hipStream_t stream) {
  const float* beta        = (const float*)d_in[0];  // 128 x 10
  const float* theta       = (const float*)d_in[1];  // 128 x 72
  const float* v_template  = (const float*)d_in[2];  // 6890 x 3
  const float* shapedirs   = (const float*)d_in[3];  // 10 x 20670
  const float* posedirs    = (const float*)d_in[4];  // 207 x 20670
  const float* J_regressor = (const float*)d_in[5];  // 24 x 6890
  const float* joint_reg   = (const float*)d_in[6];  // 43 x 6890
  const float* weights     = (const float*)d_in[7];  // 6890 x 24

  float* verts  = (float*)d_out;                     // B*V*3
  float* joints = verts + (size_t)BATCH * NV * 3;    // B*43*3

  float* ws = (float*)d_ws;
  float* Rs       = ws;                               // 128*24*9  = 27648
  float* pf       = Rs + BATCH * NJ * 9;              // 128*208   = 26624
  float* v_shaped = pf + BATCH * NPP;                 // 128*20670
  float* v_posed  = v_shaped + (size_t)BATCH * NTOT;  // 128*20670
  float* Jw       = v_posed + (size_t)BATCH * NTOT;   // 128*24*3
  float* Aw       = Jw + BATCH * NJ * 3;              // 128*24*16

  // K1: rotations + pose feature (padded stride 208)
  rodrigues_kernel<<<(BATCH * NJ + 255) / 256, 256, 0, stream>>>(theta, Rs, pf);

  // K2: fused blend GEMM (WMMA fp32)
  dim3 g2((NTOT + 15) / 16, BATCH / 16);
  blend_gemm_kernel<<<g2, 32, 0, stream>>>(beta, pf, shapedirs, posedirs,
                                           v_template, v_shaped, v_posed);

  // K3: J = J_regressor . v_shaped
  regress_kernel<<<BATCH * NJ, 256, 0, stream>>>(J_regressor, v_shaped, Jw, NJ);

  // K4: kinematic chain -> A (B x 24 x 16)
  chain_kernel<<<1, 128, 0, stream>>>(Rs, Jw, Aw);

  // K5: skinning (WMMA fp32) -> verts
  dim3 g5((NV + 15) / 16, BATCH);
  skin_kernel<<<g5, 32, 0, stream>>>(weights, Aw, v_posed, verts);

  // K6: joints = joint_regressor . verts
  regress_kernel<<<BATCH * JOUT, 256, 0, stream>>>(joint_reg, verts, joints,
                                                   JOUT);
}